// CTCLoss_27187142984058
// MI455X (gfx1250) — compile-verified
//
#include <hip/hip_runtime.h>
#include <math.h>

// Problem constants (fixed by the reference).
#define T_   256
#define N_   64
#define C_   4096
#define L_   32
#define S_   65          // 2*L+1
#define NEGV (-1e30f)

typedef unsigned int u32;
typedef u32 u32x4 __attribute__((ext_vector_type(4)));
typedef int  i32x8 __attribute__((ext_vector_type(8)));
typedef int  i32x4 __attribute__((ext_vector_type(4)));

// ---------------- wave32 reduction helpers ----------------
__device__ __forceinline__ float wave_max(float v) {
  #pragma unroll
  for (int off = 16; off > 0; off >>= 1) v = fmaxf(v, __shfl_xor(v, off, 32));
  return v;
}
__device__ __forceinline__ float wave_sum(float v) {
  #pragma unroll
  for (int off = 16; off > 0; off >>= 1) v += __shfl_xor(v, off, 32);
  return v;
}

// logaddexp, safe for NEGV sentinels (exp underflows to 0)
__device__ __forceinline__ float lae(float x, float y) {
  float m = fmaxf(x, y);
  float d = fminf(x, y) - m;           // <= 0, finite
  return m + log1pf(__expf(d));
}

// =====================================================================
// Phase 1: per-(t,n) logsumexp over C via TDM row DMA into LDS,
// then gather lp_ext[t][n][s] = preds[t,n,ext[s]] - lse into workspace.
// One block per (t,n) row; 256 threads (8 waves).
// =====================================================================
__global__ __launch_bounds__(256) void ctc_rowlse_kernel(
    const float* __restrict__ preds, const int* __restrict__ targets,
    float* __restrict__ lp) {
  __shared__ float sbuf[C_];     // 16 KB row staged by the Tensor Data Mover
  __shared__ float wred[8];
  __shared__ float sbc[2];

  const int tid = threadIdx.x;
  const int b   = blockIdx.x;          // b = t*N + n ; preds is [T][N][C]
  const int n   = b % N_;
  const float* row = preds + (size_t)b * C_;

#if __has_builtin(__builtin_amdgcn_tensor_load_to_lds)
  if (tid < 32) {                      // wave 0 issues one 16 KB DMA
    unsigned long long ga = (unsigned long long)(uintptr_t)row;
    u32 lds_off = (u32)(uintptr_t)sbuf;      // flat LDS ptr: low 32 bits = LDS byte offset

    // D# group 0: count=1 | lds_addr | global_addr[56:0] | type=2
    u32x4 g0;
    g0[0] = 1u;
    g0[1] = lds_off;
    g0[2] = (u32)ga;
    g0[3] = ((u32)(ga >> 32) & 0x1FFFFFFu) | (2u << 30);

    // D# group 1: data_size=4B; tensor_dim0=tile_dim0=4096; dim1 sizes = 1;
    // tensor_dim0_stride = 4096 elements.
    i32x8 g1;
    g1[0] = (int)(2u << 16);                         // wg_mask=0, data_size=2 (4B)
    g1[1] = (int)(((u32)C_ & 0xFFFFu) << 16);        // tensor_dim0[15:0]
    g1[2] = (int)(1u << 16);                         // tensor_dim0 hi=0, tensor_dim1=1
    g1[3] = (int)(((u32)C_ & 0xFFFFu) << 16);        // tile_dim0 = 4096
    g1[4] = (int)(1u | (1u << 16));                  // tile_dim1=1, tile_dim2=1
    g1[5] = C_;                                      // tensor_dim0_stride
    g1[6] = (int)(((u32)C_ & 0xFFFFu) << 16);        // tensor_dim1_stride (unused row)
    g1[7] = 0;

    // D# groups 2/3: higher dims collapsed to size/tile 1, strides 0.
    i32x4 g2; g2[0] = 1; g2[1] = 1; g2[2] = 0; g2[3] = (int)(1u << 16);
    i32x4 g3; g3[0] = 0; g3[1] = (int)(1u << 16); g3[2] = (int)(1u << 16); g3[3] = 0;

    // Extra group required by the clang-23 6-arg builtin; zero-filled.
    i32x8 g4;
    g4[0] = 0; g4[1] = 0; g4[2] = 0; g4[3] = 0;
    g4[4] = 0; g4[5] = 0; g4[6] = 0; g4[7] = 0;

    __builtin_amdgcn_tensor_load_to_lds(g0, g1, g2, g3, g4, 0);
#if __has_builtin(__builtin_amdgcn_s_wait_tensorcnt)
    __builtin_amdgcn_s_wait_tensorcnt(0);
#else
    asm volatile("s_wait_tensorcnt 0" ::: "memory");
#endif
  }
#else
  // Fallback: cooperative vector loads
  {
    const float4* g4v = (const float4*)row;
    float4* s4w = (float4*)sbuf;
    #pragma unroll
    for (int k = 0; k < 4; ++k) s4w[tid + k * 256] = g4v[tid + k * 256];
  }
#endif
  __syncthreads();

  // Register-resident 16 elements/thread, two-pass logsumexp.
  const float4* s4 = (const float4*)sbuf;
  float4 r[4];
  #pragma unroll
  for (int k = 0; k < 4; ++k) r[k] = s4[tid + k * 256];

  float m = fmaxf(fmaxf(r[0].x, r[0].y), fmaxf(r[0].z, r[0].w));
  #pragma unroll
  for (int k = 1; k < 4; ++k)
    m = fmaxf(m, fmaxf(fmaxf(r[k].x, r[k].y), fmaxf(r[k].z, r[k].w)));

  const int wid = tid >> 5, lane = tid & 31;
  m = wave_max(m);
  if (lane == 0) wred[wid] = m;
  __syncthreads();
  if (tid < 8) {
    float v = wred[tid];
    #pragma unroll
    for (int off = 4; off > 0; off >>= 1) v = fmaxf(v, __shfl_xor(v, off, 32));
    if (tid == 0) sbc[0] = v;
  }
  __syncthreads();
  m = sbc[0];

  float ssum = 0.f;
  #pragma unroll
  for (int k = 0; k < 4; ++k) {
    ssum += __expf(r[k].x - m); ssum += __expf(r[k].y - m);
    ssum += __expf(r[k].z - m); ssum += __expf(r[k].w - m);
  }
  ssum = wave_sum(ssum);
  if (lane == 0) wred[wid] = ssum;
  __syncthreads();
  if (tid < 8) {
    float v = wred[tid];
    #pragma unroll
    for (int off = 4; off > 0; off >>= 1) v += __shfl_xor(v, off, 32);
    if (tid == 0) sbc[1] = m + __logf(v);
  }
  __syncthreads();
  const float lse = sbc[1];

  // Gather extended-label log-probs: s even -> blank(0), s odd -> targets[n][s>>1]
  if (tid < S_) {
    int lbl = (tid & 1) ? targets[n * L_ + (tid >> 1)] : 0;
    lp[(size_t)b * S_ + tid] = sbuf[lbl] - lse;
  }
}

// =====================================================================
// Phase 2: CTC alpha recursion. One wave32 per batch element; lane l owns
// states 3l..3l+2; neighbors via __shfl_up -> zero barriers in the T loop.
// =====================================================================
__global__ __launch_bounds__(32) void ctc_alpha_kernel(
    const float* __restrict__ lp, const int* __restrict__ targets,
    const int* __restrict__ plen, const int* __restrict__ tlen,
    float* __restrict__ nll_out) {
  __shared__ float alpha_sh[S_];
  const int n    = blockIdx.x;
  const int lane = threadIdx.x;
  const int tl   = tlen[n];
  const int pl   = plen[n];
  const int s0   = 3 * lane;

  bool valid[3], skip[3];
  #pragma unroll
  for (int j = 0; j < 3; ++j) {
    int s = s0 + j;
    bool inr = (s < S_);
    valid[j] = inr && (s < 2 * tl + 1);
    bool sk = false;
    if (inr && s >= 2 && (s & 1)) {           // even states are blank -> never skip
      int lb  = targets[n * L_ + (s >> 1)];
      int lb2 = targets[n * L_ + (s >> 1) - 1];
      sk = (lb != 0) && (lb != lb2);
    }
    skip[j] = sk;
  }

  // alpha at t=0
  float a[3];
  #pragma unroll
  for (int j = 0; j < 3; ++j) {
    int s = s0 + j;
    float v = NEGV;
    if (s < S_ && valid[j] && s <= 1) v = lp[(size_t)n * S_ + s];
    a[j] = v;
  }

  for (int t = 1; t < T_; ++t) {
    const float* lpt = lp + ((size_t)t * N_ + n) * S_;
    float l0 = (s0     < S_) ? lpt[s0]     : 0.f;
    float l1 = (s0 + 1 < S_) ? lpt[s0 + 1] : 0.f;
    float l2 = (s0 + 2 < S_) ? lpt[s0 + 2] : 0.f;
    if (t + 1 < T_)
      __builtin_prefetch(lp + ((size_t)(t + 1) * N_ + n) * S_ + s0, 0, 0);

    float pa2 = __shfl_up(a[2], 1, 32);
    float pa1 = __shfl_up(a[1], 1, 32);
    if (lane == 0) { pa2 = NEGV; pa1 = NEGV; }

    float n0 = lae(lae(a[0], pa2 ), skip[0] ? pa1  : NEGV);
    float n1 = lae(lae(a[1], a[0]), skip[1] ? pa2  : NEGV);
    float n2 = lae(lae(a[2], a[1]), skip[2] ? a[0] : NEGV);
    n0 = valid[0] ? n0 + l0 : NEGV;
    n1 = valid[1] ? n1 + l1 : NEGV;
    n2 = valid[2] ? n2 + l2 : NEGV;

    bool act = (t < pl);
    a[0] = act ? n0 : a[0];
    a[1] = act ? n1 : a[1];
    a[2] = act ? n2 : a[2];
  }

  #pragma unroll
  for (int j = 0; j < 3; ++j)
    if (s0 + j < S_) alpha_sh[s0 + j] = a[j];
  __syncthreads();

  if (lane == 0) {
    int end = 2 * tl;
    float al  = alpha_sh[end];
    float ap  = (tl > 0) ? alpha_sh[end - 1] : NEGV;
    float nll = -lae(al, ap);
    if (!(isfinite(nll) && nll < 1e29f)) nll = 0.f;
    int d = (tl > 1) ? tl : 1;
    nll_out[n] = nll / (float)d;
  }
}

// =====================================================================
// Phase 3: mean over N=64 -> scalar.
// =====================================================================
__global__ __launch_bounds__(32) void ctc_mean_kernel(
    const float* __restrict__ nll, float* __restrict__ out) {
  int lane = threadIdx.x;
  float v = nll[lane] + nll[lane + 32];
  v = wave_sum(v);
  if (lane == 0) out[0] = v / (float)N_;
}

extern "C" void kernel_launch(void* const* d_in, const int* in_sizes, int n_in,
                              void* d_out, int out_size, void* d_ws, size_t ws_size,
                              hipStream_t stream) {
  const float* preds   = (const float*)d_in[0];
  const int*   targets = (const int*)d_in[1];
  const int*   plen    = (const int*)d_in[2];
  const int*   tlen    = (const int*)d_in[3];

  float* lp  = (float*)d_ws;                     // T*N*S floats (~4.26 MB)
  float* nll = lp + (size_t)T_ * N_ * S_;        // N floats

  ctc_rowlse_kernel<<<T_ * N_, 256, 0, stream>>>(preds, targets, lp);
  ctc_alpha_kernel<<<N_, 32, 0, stream>>>(lp, targets, plen, tlen, nll);
  ctc_mean_kernel<<<1, 32, 0, stream>>>(nll, (float*)d_out);
}